// SpaceChargeKick_87162066305813
// MI455X (gfx1250) — compile-verified
//
#include <hip/hip_runtime.h>
#include <math.h>

#define C_D     299792458.0
#define EPS0_D  8.8541878128e-12
#define ME_D    9.1093837015e-31
#define QE_D    1.602176634e-19
#define PI_D    3.14159265358979323846

#define NXG   64
#define NDG   128
#define NCELL (NXG*NXG*NXG)
#define NGRID (NDG*NDG*NDG)
#define NPART 1000000

typedef float v2f __attribute__((ext_vector_type(2)));
typedef float v8f __attribute__((ext_vector_type(8)));
typedef unsigned int u32x4 __attribute__((ext_vector_type(4)));
typedef int i32x4_t __attribute__((ext_vector_type(4)));
typedef int i32x8_t __attribute__((ext_vector_type(8)));

struct Params {
  double gammaref, betaref, p0, kickScale;
  double gdx, gdy, gdt;      // grid half-dimensions
  double cx, cy, cz;         // cell sizes
  double inv_cv;             // 1/cell volume
  double dxg, dyg, dsg;      // green-fn cell sizes (dsg includes *gammaref)
  double escx, escy, escz;   // E-field scale: -1/gamma^2 * 0.5/cell / (4 pi eps0)
};

// ---------------- clear scratch ----------------
__global__ __launch_bounds__(256) void k_clear(float* __restrict__ rho, double* __restrict__ acc) {
  int i = blockIdx.x * 256 + threadIdx.x;
  if (i < NCELL) rho[i] = 0.f;
  if (i < 6) acc[i] = 0.0;
}

// ---------------- moment reduction (mean/var of x,y,s) ----------------
__global__ __launch_bounds__(256) void k_reduce(const float* __restrict__ parts, double* __restrict__ acc) {
  __shared__ double sm[6 * 256];
  double v0 = 0, v1 = 0, v2 = 0, v3 = 0, v4 = 0, v5 = 0;
  for (int i = blockIdx.x * 256 + threadIdx.x; i < NPART; i += gridDim.x * 256) {
    double x = parts[i * 7 + 0], y = parts[i * 7 + 2], s = parts[i * 7 + 4];
    v0 += x; v1 += y; v2 += s;
    v3 += x * x; v4 += y * y; v5 += s * s;
  }
  sm[0 * 256 + threadIdx.x] = v0; sm[1 * 256 + threadIdx.x] = v1; sm[2 * 256 + threadIdx.x] = v2;
  sm[3 * 256 + threadIdx.x] = v3; sm[4 * 256 + threadIdx.x] = v4; sm[5 * 256 + threadIdx.x] = v5;
  __syncthreads();
  for (int st = 128; st > 0; st >>= 1) {
    if (threadIdx.x < st)
      for (int c = 0; c < 6; ++c) sm[c * 256 + threadIdx.x] += sm[c * 256 + threadIdx.x + st];
    __syncthreads();
  }
  if (threadIdx.x == 0)
    for (int c = 0; c < 6; ++c) atomicAdd(&acc[c], sm[c * 256]);
}

// ---------------- derived scalars ----------------
__global__ void k_params(const float* __restrict__ energy, const double* __restrict__ acc, Params* P) {
  if (threadIdx.x || blockIdx.x) return;
  const double N = (double)NPART;
  double E0 = ME_D * C_D * C_D / QE_D;               // rest energy [eV]
  double gref = (double)energy[0] / E0;
  double bref = sqrt(1.0 - 1.0 / (gref * gref));
  double dt = 1.0 / (C_D * bref);                    // LENGTH = 1
  double p0 = gref * bref * ME_D * C_D;
  double vx = (acc[3] - acc[0] * acc[0] / N) / (N - 1.0);
  double vy = (acc[4] - acc[1] * acc[1] / N) / (N - 1.0);
  double vs = (acc[5] - acc[2] * acc[2] / N) / (N - 1.0);
  double gdx = 3.0 * sqrt(vx), gdy = 3.0 * sqrt(vy), gdt = 3.0 * bref * sqrt(vs);
  double cx = gdx / 32.0, cy = gdy / 32.0, cz = gdt / 32.0;   // 2*gd/64
  double ig2 = 1.0 / (gref * gref);
  double phiScale = 1.0 / (4.0 * PI_D * EPS0_D);
  P->gammaref = gref; P->betaref = bref; P->p0 = p0; P->kickScale = QE_D * dt;
  P->gdx = gdx; P->gdy = gdy; P->gdt = gdt;
  P->cx = cx; P->cy = cy; P->cz = cz;
  P->inv_cv = 1.0 / (cx * cy * cz);
  P->dxg = cx; P->dyg = cy; P->dsg = cz * gref;
  P->escx = -ig2 * 0.5 / cx * phiScale;
  P->escy = -ig2 * 0.5 / cy * phiScale;
  P->escz = -ig2 * 0.5 / cz * phiScale;
}

// ---------------- DFT twiddle matrices (forward & inverse) ----------------
__global__ __launch_bounds__(256) void k_genw(float* wfr, float* wfi, float* wir, float* wii) {
  int idx = blockIdx.x * 256 + threadIdx.x;
  if (idx >= NDG * NDG) return;
  int k = idx >> 7, n = idx & 127;
  int m = (k * n) & 127;
  double ang = -2.0 * PI_D * (double)m / 128.0;
  double sv, cv;
  sincos(ang, &sv, &cv);
  wfr[idx] = (float)cv; wfi[idx] = (float)sv;      // exp(-i...)
  wir[idx] = (float)cv; wii[idx] = (float)(-sv);   // exp(+i...)
}

// ---------------- CIC deposit ----------------
__global__ __launch_bounds__(256) void k_deposit(const float* __restrict__ parts,
                                                 const float* __restrict__ q,
                                                 const Params* __restrict__ P,
                                                 float* __restrict__ rho) {
  int i = blockIdx.x * 256 + threadIdx.x;
  if (i >= NPART) return;
  float x = parts[i * 7 + 0], y = parts[i * 7 + 2], s = parts[i * 7 + 4];
  float tau = -(float)P->betaref * s;
  float nx = (x + (float)P->gdx) / (float)P->cx;
  float ny = (y + (float)P->gdy) / (float)P->cy;
  float nt = (tau + (float)P->gdt) / (float)P->cz;
  int bx = (int)floorf(nx), by = (int)floorf(ny), bt = (int)floorf(nt);
  float qi = q[i];
#pragma unroll
  for (int o = 0; o < 8; ++o) {
    int gx = bx + ((o >> 2) & 1), gy = by + ((o >> 1) & 1), gz = bt + (o & 1);
    if ((unsigned)gx < NXG && (unsigned)gy < NXG && (unsigned)gz < NXG) {
      float w = (1.f - fabsf(nx - (float)gx)) * (1.f - fabsf(ny - (float)gy)) * (1.f - fabsf(nt - (float)gz));
      atomicAdd(&rho[(gx * NXG + gy) * NXG + gz], w * qi);
    }
  }
}

// ---------------- integrated Green function on doubled grid ----------------
__device__ inline double ipot(double x, double y, double s) {
  double r = sqrt(x * x + y * y + s * s);
  return -0.5 * s * s * atan(x * y / (s * r))
         - 0.5 * y * y * atan(x * s / (y * r))
         - 0.5 * x * x * atan(y * s / (x * r))
         + y * s * asinh(x / sqrt(y * y + s * s))
         + x * s * asinh(y / sqrt(x * x + s * s))
         + x * y * asinh(s / sqrt(x * x + y * y));
}

__global__ __launch_bounds__(256) void k_green(const Params* __restrict__ P,
                                               float* __restrict__ gr, float* __restrict__ gi) {
  int idx = blockIdx.x * 256 + threadIdx.x;
  if (idx >= NGRID) return;
  gi[idx] = 0.f;
  int c = idx & 127, b = (idx >> 7) & 127, a = idx >> 14;
  if (a == NXG || b == NXG || c == NXG) { gr[idx] = 0.f; return; }
  int ia = (a < NXG) ? a : (NDG - a);
  int ib = (b < NXG) ? b : (NDG - b);
  int ic = (c < NXG) ? c : (NDG - c);
  double dx = P->dxg, dy = P->dyg, ds = P->dsg;
  double xg = ia * dx, yg = ib * dy, sg = ic * ds;
  double hx = 0.5 * dx, hy = 0.5 * dy, hz = 0.5 * ds;
  double G = ipot(xg + hx, yg + hy, sg + hz) - ipot(xg - hx, yg + hy, sg + hz)
           - ipot(xg + hx, yg - hy, sg + hz) - ipot(xg + hx, yg + hy, sg - hz)
           + ipot(xg + hx, yg - hy, sg - hz) + ipot(xg - hx, yg + hy, sg - hz)
           + ipot(xg - hx, yg - hy, sg + hz) - ipot(xg - hx, yg - hy, sg - hz);
  gr[idx] = (float)G;
}

// ---------------- embed rho into doubled complex grid ----------------
__global__ __launch_bounds__(256) void k_pack(const float* __restrict__ rho, const Params* __restrict__ P,
                                              float* __restrict__ pr, float* __restrict__ pi) {
  int idx = blockIdx.x * 256 + threadIdx.x;
  if (idx >= NGRID) return;
  int c = idx & 127, b = (idx >> 7) & 127, a = idx >> 14;
  float v = 0.f;
  if (a < NXG && b < NXG && c < NXG) v = rho[(a * NXG + b) * NXG + c] * (float)P->inv_cv;
  pr[idx] = v;
  pi[idx] = 0.f;
}

// ---------------- TDM: DMA a 128x32 f32 tile (row stride 128) into LDS ----------------
__device__ inline unsigned ldsByteOffset(const void* p) {
  return (unsigned)(uintptr_t)(__attribute__((address_space(3))) const void*)p;
}

#if __has_builtin(__builtin_amdgcn_tensor_load_to_lds)
#define HAVE_TDM 1
__device__ inline void tdmLoadW(const float* g, unsigned ldsOff) {
  unsigned long long ga = (unsigned long long)(uintptr_t)g;
  // D# group 0: count=1 | lds_addr | global_addr[56:0] | type=2
  u32x4 g0 = { 1u,
               ldsOff,
               (unsigned)(ga & 0xFFFFFFFFu),
               (unsigned)((ga >> 32) & 0x1FFFFFFu) | (2u << 30) };
  // D# group 1: data_size=4B; tensor 128x128; tile 32x128; strides 128 / 16384
  i32x8_t g1 = { (int)0x00020000,            // mask=0, data_size=2 (4B)
                 (int)(128u << 16),          // tensor_dim0[15:0] in bits[63:48]
                 (int)(128u << 16),          // dim0 hi=0, tensor_dim1[15:0]
                 (int)(32u << 16),           // dim1 hi=0, tile_dim0=32
                 (int)128,                   // tile_dim1=128, tile_dim2=0
                 (int)128,                   // tensor_dim0_stride lo
                 (int)0x40000000,            // stride0 hi=0, stride1 lo=16384
                 (int)0 };                   // stride1 hi
  i32x4_t z4 = { 0, 0, 0, 0 };
  i32x8_t z8 = { 0, 0, 0, 0, 0, 0, 0, 0 };
  __builtin_amdgcn_tensor_load_to_lds(g0, g1, z4, z4, z8, 0);
}
#else
#define HAVE_TDM 0
#endif

// ---------------- 1D complex DFT along one axis as WMMA f32 GEMM ----------------
// Out[v,n] = sum_t X[v,t] * W[t,n]  (W symmetric DFT matrix).
// element offset: (v>>7)*strideHi + (v&127)*strideLo + t*sT
// grid(256, 4), block(128) = 4 waves; each wave owns one 16(v) x 32(n) complex tile.
__global__ __launch_bounds__(128) void k_dft(
    const float* __restrict__ srcR, const float* __restrict__ srcI,
    float* __restrict__ dstR, float* __restrict__ dstI,
    const float* __restrict__ wR, const float* __restrict__ wI,
    int strideLo, int strideHi, int sT) {
  __shared__ float ldsWr[128 * 32];
  __shared__ float ldsWi[128 * 32];
  const int tid = threadIdx.x;
  const int lane = tid & 31;
  const int wave = tid >> 5;
  const int n0 = blockIdx.y * 32;

#if HAVE_TDM
  if (tid == 0) {
    tdmLoadW(wR + n0, ldsByteOffset(ldsWr));
    tdmLoadW(wI + n0, ldsByteOffset(ldsWi));
    __builtin_amdgcn_s_wait_tensorcnt(0);
  }
#else
  for (int r = tid; r < 128 * 32; r += 128) {
    int t = r >> 5, nl = r & 31;
    ldsWr[r] = wR[t * 128 + n0 + nl];
    ldsWi[r] = wI[t * 128 + n0 + nl];
  }
#endif
  __syncthreads();

  const int v0 = (blockIdx.x * 4 + wave) * 16;
  const int rowBase = (v0 >> 7) * strideHi + (v0 & 127) * strideLo;

  const int mA = lane & 15;          // A-fragment row (M)
  const int tOff = (lane >> 4) * 2;  // K offset: lanes 16-31 hold K+2
  const int nl = lane & 15;          // B-fragment column

  v8f accR0 = {0.f, 0.f, 0.f, 0.f, 0.f, 0.f, 0.f, 0.f};
  v8f accI0 = accR0, accR1 = accR0, accI1 = accR0;

  for (int k0 = 0; k0 < 128; k0 += 4) {
    int aBase = rowBase + mA * strideLo + (k0 + tOff) * sT;
    v2f ar, ai, ani;
    ar.x = srcR[aBase]; ar.y = srcR[aBase + sT];
    ai.x = srcI[aBase]; ai.y = srcI[aBase + sT];
    ani = -ai;
    int b0 = (k0 + tOff) * 32 + nl;
    v2f br0, bi0, br1, bi1;
    br0.x = ldsWr[b0];      br0.y = ldsWr[b0 + 32];
    bi0.x = ldsWi[b0];      bi0.y = ldsWi[b0 + 32];
    br1.x = ldsWr[b0 + 16]; br1.y = ldsWr[b0 + 48];
    bi1.x = ldsWi[b0 + 16]; bi1.y = ldsWi[b0 + 48];

    accR0 = __builtin_amdgcn_wmma_f32_16x16x4_f32(false, ar,  false, br0, (short)0, accR0, false, false);
    accR0 = __builtin_amdgcn_wmma_f32_16x16x4_f32(false, ani, false, bi0, (short)0, accR0, false, false);
    accI0 = __builtin_amdgcn_wmma_f32_16x16x4_f32(false, ar,  false, bi0, (short)0, accI0, false, false);
    accI0 = __builtin_amdgcn_wmma_f32_16x16x4_f32(false, ai,  false, br0, (short)0, accI0, false, false);
    accR1 = __builtin_amdgcn_wmma_f32_16x16x4_f32(false, ar,  false, br1, (short)0, accR1, false, false);
    accR1 = __builtin_amdgcn_wmma_f32_16x16x4_f32(false, ani, false, bi1, (short)0, accR1, false, false);
    accI1 = __builtin_amdgcn_wmma_f32_16x16x4_f32(false, ar,  false, bi1, (short)0, accI1, false, false);
    accI1 = __builtin_amdgcn_wmma_f32_16x16x4_f32(false, ai,  false, br1, (short)0, accI1, false, false);
  }

  // C/D layout: VGPR r -> lanes 0-15: M=r; lanes 16-31: M=r+8; N=lane&15
  const int mBase = (lane >> 4) * 8;
#pragma unroll
  for (int r = 0; r < 8; ++r) {
    int m = mBase + r;
    int off0 = rowBase + m * strideLo + (n0 + nl) * sT;
    int off1 = off0 + 16 * sT;
    dstR[off0] = accR0[r];
    dstI[off0] = accI0[r];
    dstR[off1] = accR1[r];
    dstI[off1] = accI1[r];
  }
}

// ---------------- spectral multiply (folds 1/128^3 ifft scale) ----------------
__global__ __launch_bounds__(256) void k_mul(const float* __restrict__ ar, const float* __restrict__ ai,
                                             const float* __restrict__ br, const float* __restrict__ bi,
                                             float* __restrict__ dr, float* __restrict__ di) {
  int i = blockIdx.x * 256 + threadIdx.x;
  if (i >= NGRID) return;
  const float sc = 1.0f / 2097152.0f;
  float a = ar[i], b = ai[i], c = br[i], d = bi[i];
  dr[i] = (a * c - b * d) * sc;
  di[i] = (a * d + b * c) * sc;
}

// ---------------- E-field: central differences with zero boundary ----------------
__global__ __launch_bounds__(256) void k_field(const Params* __restrict__ P, const float* __restrict__ phi,
                                               float* __restrict__ Ex, float* __restrict__ Ey, float* __restrict__ Es) {
  int idx = blockIdx.x * 256 + threadIdx.x;
  if (idx >= NCELL) return;
  int iz = idx & 63, iy = (idx >> 6) & 63, ix = idx >> 12;
#define PHI(A, B, Cc) phi[(((A) * NDG) + (B)) * NDG + (Cc)]
  float ex = 0.f, ey = 0.f, es = 0.f;
  if (ix >= 1 && ix <= 62) ex = (PHI(ix + 1, iy, iz) - PHI(ix - 1, iy, iz)) * (float)P->escx;
  if (iy >= 1 && iy <= 62) ey = (PHI(ix, iy + 1, iz) - PHI(ix, iy - 1, iz)) * (float)P->escy;
  if (iz >= 1 && iz <= 62) es = (PHI(ix, iy, iz + 1) - PHI(ix, iy, iz - 1)) * (float)P->escz;
#undef PHI
  Ex[idx] = ex; Ey[idx] = ey; Es[idx] = es;
}

// ---------------- CIC gather + momentum kick ----------------
__global__ __launch_bounds__(256) void k_gather(const float* __restrict__ parts,
                                                const Params* __restrict__ P,
                                                const float* __restrict__ Ex,
                                                const float* __restrict__ Ey,
                                                const float* __restrict__ Es,
                                                float* __restrict__ out) {
  int i = blockIdx.x * 256 + threadIdx.x;
  if (i >= NPART) return;
  float x = parts[i * 7 + 0], xp = parts[i * 7 + 1], y = parts[i * 7 + 2], yp = parts[i * 7 + 3];
  float s = parts[i * 7 + 4], del = parts[i * 7 + 5], hom = parts[i * 7 + 6];
  double gref = P->gammaref, bref = P->betaref, p0 = P->p0;
  double px = p0 * (double)xp;
  double py = p0 * (double)yp;
  double gamma = gref * (1.0 + bref * (double)del);
  double beta = sqrt(1.0 - 1.0 / (gamma * gamma));
  double pm = gamma * ME_D * beta * C_D;
  double psq = pm * pm - px * px - py * py;
  double ps = sqrt(psq > 0.0 ? psq : 0.0);

  float tau = -(float)bref * s;
  float nx = (x + (float)P->gdx) / (float)P->cx;
  float ny = (y + (float)P->gdy) / (float)P->cy;
  float nt = (tau + (float)P->gdt) / (float)P->cz;
  int bx = (int)floorf(nx), by = (int)floorf(ny), bt = (int)floorf(nt);
  float fx = 0.f, fy = 0.f, fs = 0.f;
#pragma unroll
  for (int o = 0; o < 8; ++o) {
    int gx = bx + ((o >> 2) & 1), gy = by + ((o >> 1) & 1), gz = bt + (o & 1);
    if ((unsigned)gx < NXG && (unsigned)gy < NXG && (unsigned)gz < NXG) {
      float w = (1.f - fabsf(nx - (float)gx)) * (1.f - fabsf(ny - (float)gy)) * (1.f - fabsf(nt - (float)gz));
      int idx = (gx * NXG + gy) * NXG + gz;
      fx += w * Ex[idx]; fy += w * Ey[idx]; fs += w * Es[idx];
    }
  }
  double ks = P->kickScale;  // E_CHARGE * dt
  double px2 = px + (double)fx * ks;
  double py2 = py + (double)fy * ks;
  double ps2 = ps + (double)fs * ks;
  double pn = sqrt(px2 * px2 + py2 * py2 + ps2 * ps2);
  double t = pn / (ME_D * C_D);
  double gnew = sqrt(1.0 + t * t);
  out[i * 7 + 0] = x;
  out[i * 7 + 1] = (float)(px2 / p0);
  out[i * 7 + 2] = y;
  out[i * 7 + 3] = (float)(py2 / p0);
  out[i * 7 + 4] = s;
  out[i * 7 + 5] = (float)((gnew - gref) / (bref * gref));
  out[i * 7 + 6] = hom;
}

extern "C" void kernel_launch(void* const* d_in, const int* in_sizes, int n_in,
                              void* d_out, int out_size, void* d_ws, size_t ws_size,
                              hipStream_t stream) {
  (void)in_sizes; (void)n_in; (void)out_size; (void)ws_size;
  const float* parts  = (const float*)d_in[0];
  const float* energy = (const float*)d_in[1];
  const float* charge = (const float*)d_in[2];
  float* out = (float*)d_out;

  char* ws = (char*)d_ws;
  size_t off = 0;
  auto take = [&](size_t bytes) -> char* {
    char* p = ws + off;
    off += (bytes + 255) & ~(size_t)255;
    return p;
  };
  Params* P   = (Params*)take(sizeof(Params));
  double* acc = (double*)take(6 * sizeof(double));
  float* rho  = (float*)take(NCELL * 4);
  float* Ex   = (float*)take(NCELL * 4);
  float* Ey   = (float*)take(NCELL * 4);
  float* Es   = (float*)take(NCELL * 4);
  float* Wfr  = (float*)take(NDG * NDG * 4);
  float* Wfi  = (float*)take(NDG * NDG * 4);
  float* Wir  = (float*)take(NDG * NDG * 4);
  float* Wii  = (float*)take(NDG * NDG * 4);
  float* P0r  = (float*)take((size_t)NGRID * 4);
  float* P0i  = (float*)take((size_t)NGRID * 4);
  float* P1r  = (float*)take((size_t)NGRID * 4);
  float* P1i  = (float*)take((size_t)NGRID * 4);
  float* GSr  = (float*)take((size_t)NGRID * 4);
  float* GSi  = (float*)take((size_t)NGRID * 4);

  dim3 dgrid(256, 4), dblk(128);

  k_clear<<<(NCELL + 255) / 256, 256, 0, stream>>>(rho, acc);
  k_reduce<<<512, 256, 0, stream>>>(parts, acc);
  k_params<<<1, 1, 0, stream>>>(energy, acc, P);
  k_genw<<<(NDG * NDG + 255) / 256, 256, 0, stream>>>(Wfr, Wfi, Wir, Wii);
  k_deposit<<<(NPART + 255) / 256, 256, 0, stream>>>(parts, charge, P, rho);

  // Green function spectrum: pack -> 3 forward DFT passes (axis k, j, i)
  k_green<<<NGRID / 256, 256, 0, stream>>>(P, P0r, P0i);
  k_dft<<<dgrid, dblk, 0, stream>>>(P0r, P0i, P1r, P1i, Wfr, Wfi, 128, 16384, 1);
  k_dft<<<dgrid, dblk, 0, stream>>>(P1r, P1i, P0r, P0i, Wfr, Wfi, 1, 16384, 128);
  k_dft<<<dgrid, dblk, 0, stream>>>(P0r, P0i, GSr, GSi, Wfr, Wfi, 1, 128, 16384);

  // rho spectrum
  k_pack<<<NGRID / 256, 256, 0, stream>>>(rho, P, P0r, P0i);
  k_dft<<<dgrid, dblk, 0, stream>>>(P0r, P0i, P1r, P1i, Wfr, Wfi, 128, 16384, 1);
  k_dft<<<dgrid, dblk, 0, stream>>>(P1r, P1i, P0r, P0i, Wfr, Wfi, 1, 16384, 128);
  k_dft<<<dgrid, dblk, 0, stream>>>(P0r, P0i, P1r, P1i, Wfr, Wfi, 1, 128, 16384);

  // spectral multiply (with 1/N^3), then inverse DFT
  k_mul<<<NGRID / 256, 256, 0, stream>>>(P1r, P1i, GSr, GSi, P0r, P0i);
  k_dft<<<dgrid, dblk, 0, stream>>>(P0r, P0i, P1r, P1i, Wir, Wii, 128, 16384, 1);
  k_dft<<<dgrid, dblk, 0, stream>>>(P1r, P1i, P0r, P0i, Wir, Wii, 1, 16384, 128);
  k_dft<<<dgrid, dblk, 0, stream>>>(P0r, P0i, P1r, P1i, Wir, Wii, 1, 128, 16384);

  // fields + gather/kick
  k_field<<<(NCELL + 255) / 256, 256, 0, stream>>>(P, P1r, Ex, Ey, Es);
  k_gather<<<(NPART + 255) / 256, 256, 0, stream>>>(parts, P, Ex, Ey, Es, out);
}